// TriangulationNet_36945308680199
// MI455X (gfx1250) — compile-verified
//
#include <hip/hip_runtime.h>
#include <hip/hip_bf16.h>

// ---------------------------------------------------------------------------
// Triangulation (DLT + smallest-eigenvector of 4x4 Gram) for MI455X / gfx1250.
//
// Per wave (32 points):
//   Phase 1: each lane builds its own 4x4 A matrix, stages it in LDS (b128).
//   Phase 2: software-pipelined — 8 batched ds_load_2addr, then
//            8x V_WMMA_F32_16X16X4_F32 back-to-back with staggered dscnt
//            waits (each WMMA = 4 points' Gram matrices A^T A as the diagonal
//            4x4 blocks of a 16x16 product), then one exec-masked extraction
//            pass (asm-laundered selects -> single v_cndmask per element).
//   Phase 3: all 32 lanes run a branchless cyclic-Jacobi 4x4 symmetric eigen
//            solve (FP32 VALU, hw rcp/sqrt/rsq), pick min-eigenvalue column,
//            z = clamp(v.z / v.w), filter, store kp3d + deterministic scatter.
// ---------------------------------------------------------------------------

typedef float v2f __attribute__((ext_vector_type(2)));
typedef float v8f __attribute__((ext_vector_type(8)));

#define H_IMG 480
#define W_IMG 640

__device__ __forceinline__ void wave_lds_fence() {
  // LDS ops are in-order within a wave on CDNA5; we only need to stop the
  // compiler from reordering across the staging phases.
  __builtin_amdgcn_fence(__ATOMIC_ACQ_REL, "wavefront");
  __builtin_amdgcn_wave_barrier();
}

__global__ void TriangulationNet_zero_img(float* __restrict__ img, int n) {
  int i = blockIdx.x * blockDim.x + threadIdx.x;
  if (i < n) img[i] = 0.0f;
}

__launch_bounds__(256)
__global__ void TriangulationNet_triangulate(const float* __restrict__ T,
                                             const float* __restrict__ K0,
                                             const float* __restrict__ K1,
                                             const float* __restrict__ mconf,
                                             const float* __restrict__ kp0,
                                             const float* __restrict__ kp1,
                                             float* __restrict__ img,
                                             float* __restrict__ kp3d,
                                             int N) {
  // Per-wave LDS slices: 32 A-matrices (16 f32 each) + 32 Gram matrices.
  __shared__ float AStage[8][32 * 16];
  __shared__ float GStage[8][32 * 16];

  const int lane = threadIdx.x & 31;
  const int wid  = threadIdx.x >> 5;
  const int n    = blockIdx.x * blockDim.x + threadIdx.x;

  // Kick off point-data prefetch (global_prefetch_b8) before the uniform
  // projection-matrix math so the real loads hit warm lines.
  if (n < N) {
    __builtin_prefetch(&kp0[2 * n], 0, 0);
    __builtin_prefetch(&kp1[2 * n], 0, 0);
    __builtin_prefetch(&mconf[n], 0, 0);
  }

  // ---- uniform projection matrices: P0 = [K0|0], P1 = K1 * T[:3,:] ----
  float P0[3][4], P1[3][4];
#pragma unroll
  for (int r = 0; r < 3; ++r) {
#pragma unroll
    for (int c = 0; c < 4; ++c) {
      P0[r][c] = (c < 3) ? K0[r * 3 + c] : 0.0f;
      float acc = 0.0f;
#pragma unroll
      for (int j = 0; j < 3; ++j) acc += K1[r * 3 + j] * T[j * 4 + c];
      P1[r][c] = acc;
    }
  }

  // ---- phase 1: build my 4x4 A (row-major [k][col]) and stage it ----
  float x0 = 0.0f, y0 = 0.0f;
  {
    float x1 = 0.0f, y1 = 0.0f, cf = 0.0f;
    bool valid = (n < N);
    if (valid) {
      x0 = kp0[2 * n];
      y0 = kp0[2 * n + 1];
      x1 = kp1[2 * n];
      y1 = kp1[2 * n + 1];
      cf = mconf[n];
    }
    float A[4][4];
#pragma unroll
    for (int c = 0; c < 4; ++c) {
      A[0][c] = x0 * P0[2][c] - P0[0][c];
      A[1][c] = y0 * P0[2][c] - P0[1][c];
      A[2][c] = cf * (x1 * P1[2][c] - P1[0][c]);
      A[3][c] = cf * (y1 * P1[2][c] - P1[1][c]);
    }
    if (!valid) {
#pragma unroll
      for (int r = 0; r < 4; ++r)
#pragma unroll
        for (int c = 0; c < 4; ++c) A[r][c] = 0.0f;
    }
#pragma unroll
    for (int r = 0; r < 4; ++r)
#pragma unroll
      for (int c = 0; c < 4; ++c)
        AStage[wid][lane * 16 + r * 4 + c] = A[r][c];
  }
  wave_lds_fence();

  // ---- phase 2: batched loads -> 8 WMMAs -> extraction ----
  // A-operand 16x4 layout: lane L holds M = L&15, K = {0,1} (L<16) / {2,3}.
  // X[m][k] = A_{m>>2}[k][m&3]; B-operand Y[k][n] = A_{n>>2}[k][n&3] —
  // identical per-lane loads, so one v2f feeds both operands.
  const int Mn = lane & 15;
  const int kb = (lane < 16) ? 0 : 2;
  const int pl = Mn >> 2;   // point-within-quad from M/N coordinate
  const int ci = Mn & 3;    // column inside the 4x4 block

  // (a) issue all operand loads back-to-back; dscnt waits overlap the WMMAs
  v2f ab[8];
#pragma unroll
  for (int g = 0; g < 8; ++g) {
    const int sp = g * 4 + pl;  // which staged point this lane's operand reads
    ab[g].x = AStage[wid][sp * 16 + kb * 4 + ci];
    ab[g].y = AStage[wid][sp * 16 + (kb + 1) * 4 + ci];
  }

  // (b) 8 independent WMMAs, distinct accumulators -> no WAR serialization
  v8f acc[8];
#pragma unroll
  for (int g = 0; g < 8; ++g) {
    v8f czero = {};
    acc[g] = __builtin_amdgcn_wmma_f32_16x16x4_f32(
        /*neg_a=*/false, ab[g], /*neg_b=*/false, ab[g],
        /*c_mod=*/(short)0, czero, /*reuse_a=*/false, /*reuse_b=*/false);
  }

  // (c) extraction of the diagonal 4x4 blocks.
  // D (16x16 f32) layout: lanes 0-15: VGPR r -> (M=r, N=lane);
  // lanes 16-31: VGPR r -> (M=8+r, N=lane-16). Diagonal blocks land in:
  //   quad 0: lanes 0-3,  VGPRs 0-3   quad 1: lanes 4-7,  VGPRs 4-7
  //   quad 2: lanes 24-27,VGPRs 0-3   quad 3: lanes 28-31,VGPRs 4-7
  // The low-quartet scalars are laundered through empty inline asm so
  // InstCombine cannot fold select(c, extract(v,4+k), extract(v,k)) into a
  // dynamic-index extract (which lowers to a cmp/cndmask cascade).
  const bool xact = (lane < 8) || (lane >= 24);
  const bool vhi  = (lane & 4) != 0;                       // VGPR quartet
  const int  grp  = ((lane >> 2) & 1) + ((lane >= 16) ? 2 : 0);
  if (xact) {
    const int j    = lane & 3;          // column of the Gram this lane holds
    const int base = grp * 16 + j;      // GStage offset for tile's point 'grp'
#pragma unroll
    for (int g = 0; g < 8; ++g) {
      float lo0 = acc[g][0], lo1 = acc[g][1];
      float lo2 = acc[g][2], lo3 = acc[g][3];
      asm("" : "+v"(lo0), "+v"(lo1), "+v"(lo2), "+v"(lo3));
      float t0 = vhi ? acc[g][4] : lo0;   // one v_cndmask each
      float t1 = vhi ? acc[g][5] : lo1;
      float t2 = vhi ? acc[g][6] : lo2;
      float t3 = vhi ? acc[g][7] : lo3;
      const int o = g * 64 + base;        // point (g*4+grp), 16 floats apart
      GStage[wid][o + 0]  = t0;
      GStage[wid][o + 4]  = t1;
      GStage[wid][o + 8]  = t2;
      GStage[wid][o + 12] = t3;
    }
  }
  wave_lds_fence();

  // ---- phase 3: per-lane branchless cyclic Jacobi on my own Gram ----
  float g4[4][4], V[4][4];
#pragma unroll
  for (int r = 0; r < 4; ++r)
#pragma unroll
    for (int c = 0; c < 4; ++c) {
      g4[r][c] = GStage[wid][lane * 16 + r * 4 + c];
      V[r][c]  = (r == c) ? 1.0f : 0.0f;
    }

  // Branchless rotation: when |apq| is tiny, force t = 0 -> c = 1, s = 0
  // (identity update) instead of exec-mask divergence. Raw hw v_rcp/v_sqrt/
  // v_rsq (1-ulp) are plenty inside an iterative eigen solve.
#define JROT(p_, q_)                                                         \
  {                                                                          \
    float apq  = g4[p_][q_];                                                 \
    bool  tiny = (fabsf(apq) <= 1e-20f);                                     \
    float rden = __builtin_amdgcn_rcpf(tiny ? 1.0f : (2.0f * apq));          \
    float tau  = (g4[q_][q_] - g4[p_][p_]) * rden;                           \
    float t    = ((tau >= 0.0f) ? 1.0f : -1.0f) *                            \
                 __builtin_amdgcn_rcpf(                                      \
                     fabsf(tau) +                                            \
                     __builtin_amdgcn_sqrtf(fmaf(tau, tau, 1.0f)));          \
    t = tiny ? 0.0f : t;                                                     \
    float c = __builtin_amdgcn_rsqf(fmaf(t, t, 1.0f));                       \
    float s = t * c;                                                         \
    _Pragma("unroll") for (int k = 0; k < 4; ++k) {                          \
      float gkp = g4[k][p_], gkq = g4[k][q_];                                \
      g4[k][p_] = fmaf(c, gkp, -s * gkq);                                    \
      g4[k][q_] = fmaf(s, gkp, c * gkq);                                     \
    }                                                                        \
    _Pragma("unroll") for (int k = 0; k < 4; ++k) {                          \
      float gpk = g4[p_][k], gqk = g4[q_][k];                                \
      g4[p_][k] = fmaf(c, gpk, -s * gqk);                                    \
      g4[q_][k] = fmaf(s, gpk, c * gqk);                                     \
    }                                                                        \
    _Pragma("unroll") for (int k = 0; k < 4; ++k) {                          \
      float vkp = V[k][p_], vkq = V[k][q_];                                  \
      V[k][p_] = fmaf(c, vkp, -s * vkq);                                     \
      V[k][q_] = fmaf(s, vkp, c * vkq);                                      \
    }                                                                        \
  }

#pragma unroll 1
  for (int sweep = 0; sweep < 6; ++sweep) {
    JROT(0, 1) JROT(0, 2) JROT(0, 3) JROT(1, 2) JROT(1, 3) JROT(2, 3)
  }
#undef JROT

  // Branchless min-eigenvalue column selection (v_cndmask chains).
  float e1 = g4[1][1], e2 = g4[2][2], e3 = g4[3][3];
  float ebest = g4[0][0];
  float vz = V[2][0], vw = V[3][0];
  {
    bool b = (e1 < ebest);
    ebest = b ? e1 : ebest; vz = b ? V[2][1] : vz; vw = b ? V[3][1] : vw;
  }
  {
    bool b = (e2 < ebest);
    ebest = b ? e2 : ebest; vz = b ? V[2][2] : vz; vw = b ? V[3][2] : vw;
  }
  {
    bool b = (e3 < ebest);
    ebest = b ? e3 : ebest; vz = b ? V[2][3] : vz; vw = b ? V[3][3] : vw;
  }

  float pz = vz / vw;                              // IEEE div: homog -> z
  pz = fminf(fmaxf(pz, -1000.0f), 1000.0f);        // reference's inf clip
  float z  = fminf(fmaxf(pz, 0.0f), 30.0f);        // depth clamp
  float kp = ((z > 0.0f) && (z < 30.0f)) ? z : 0.0f;

  if (n < N) {
    kp3d[n] = kp;
    int xi = (int)x0;   // trunc like .astype(int32)
    int yi = (int)y0;
    if (xi >= 0 && xi < W_IMG && yi >= 0 && yi < H_IMG) {
      // kp >= 0 so IEEE order == unsigned-int order: deterministic scatter.
      atomicMax((unsigned int*)&img[yi * W_IMG + xi], __float_as_uint(kp));
    }
  }
}

extern "C" void kernel_launch(void* const* d_in, const int* in_sizes, int n_in,
                              void* d_out, int out_size, void* d_ws, size_t ws_size,
                              hipStream_t stream) {
  // setup_inputs order: T_0to1, K0, K1, mconf, mkpts0_f, mkpts1_f, image0, m_bids
  const float* T     = (const float*)d_in[0];
  const float* K0    = (const float*)d_in[1];
  const float* K1    = (const float*)d_in[2];
  const float* mconf = (const float*)d_in[3];
  const float* kp0   = (const float*)d_in[4];
  const float* kp1   = (const float*)d_in[5];
  const int N = in_sizes[3];

  float* out  = (float*)d_out;
  float* img  = out;                       // [1, 480, 640]
  float* kp3d = out + H_IMG * W_IMG;       // [N]

  const int npix = H_IMG * W_IMG;
  TriangulationNet_zero_img<<<(npix + 255) / 256, 256, 0, stream>>>(img, npix);

  const int nblocks = (N + 255) / 256;
  TriangulationNet_triangulate<<<nblocks, 256, 0, stream>>>(
      T, K0, K1, mconf, kp0, kp1, img, kp3d, N);
}